// RelDConsistencyLoss_58926951301593
// MI455X (gfx1250) — compile-verified
//
#include <hip/hip_runtime.h>
#include <cstdint>

// Problem constants (match reference)
#define B_   32
#define H_   512
#define W_   512
#define P_   8192
#define N_   (H_ * W_)        // 262144 elements per batch image
#define NT_  (B_ * N_)        // 8388608 total elements
#define S_   (B_ * P_)        // 262144 sample pairs
#define EPS_ 1e-5f

// Count/compact tiling: chunks never straddle a batch (N_ % CHUNK == 0)
#define CHUNK   4096
#define NCHUNK  (NT_ / CHUNK)     // 2048 chunks (64 per batch)
#define CT      256               // threads per count/compact block
#define EPT     (CHUNK / CT)      // 16 elements per thread

// Sample-phase tiling
#define SB 256                    // blocks
#define ST 256                    // threads
#define SI (S_ / (SB * ST))       // 4 samples per thread; 1024 samples/block -> 8 blocks per batch

typedef uint32_t u32x4 __attribute__((ext_vector_type(4)));
typedef uint32_t u32x8 __attribute__((ext_vector_type(8)));

// ---------------------------------------------------------------------------
// CDNA5 async global->LDS copy (ASYNCcnt path). Inline asm: the clang builtin
// arity differs between toolchains; the ISA mnemonic does not.
// ---------------------------------------------------------------------------
__device__ __forceinline__ void async_copy_b128_to_lds(uint32_t lds_byte_addr,
                                                       uint64_t global_addr) {
    asm volatile("global_load_async_to_lds_b128 %0, %1, off"
                 :
                 : "v"(lds_byte_addr), "v"(global_addr)
                 : "memory");
}
__device__ __forceinline__ void wait_asynccnt0() {
    asm volatile("s_wait_asynccnt 0" ::: "memory");
}

// ---------------------------------------------------------------------------
// K1: per-chunk popcount of mask, chunk staged into LDS by the Tensor Data
// Mover (one DMA per block, TENSORcnt-tracked), then counted from LDS with
// bank-conflict-free strided reads.
// ---------------------------------------------------------------------------
__global__ void k_count(const int* __restrict__ mask, int* __restrict__ chunk_cnt) {
    const int blk = blockIdx.x;
    const int t   = threadIdx.x;

    __shared__ int smask[CHUNK];               // 16 KB staged mask chunk
    __shared__ int sc[CT];

    if (t == 0) {
        // Build Tensor DMA Descriptor (D#): 1-D contiguous tile of 4096 dwords.
        const uint32_t lds = (uint32_t)(uintptr_t)(&smask[0]);  // LDS byte addr
        const uint64_t ga  = (uint64_t)(uintptr_t)(mask + (size_t)blk * CHUNK);
        u32x4 g0;
        g0[0] = 1u;                                             // count=1, no gather
        g0[1] = lds;                                            // lds_addr
        g0[2] = (uint32_t)(ga & 0xFFFFFFFFu);                   // global_addr[31:0]
        g0[3] = (uint32_t)((ga >> 32) & 0x1FFFFFFu)             // global_addr[56:32]
              | (2u << 30);                                     // type=2 ("image")
        u32x8 g1;
        g1[0] = (2u << 16);                                     // data_size=4B, wg_mask=0
        g1[1] = ((uint32_t)CHUNK & 0xFFFFu) << 16;              // tensor_dim0[15:0]
        g1[2] = (1u << 16);                                     // tensor_dim0[31:16]=0, tensor_dim1=1
        g1[3] = ((uint32_t)CHUNK & 0xFFFFu) << 16;              // tensor_dim1 hi=0, tile_dim0=4096
        g1[4] = 0u;                                             // tile_dim1=0, tile_dim2=0
        g1[5] = (uint32_t)CHUNK;                                // tensor_dim0_stride[31:0]
        g1[6] = 0u;                                             // stride hi, dim1_stride lo
        g1[7] = 0u;                                             // dim1_stride hi
        asm volatile("tensor_load_to_lds %0, %1"
                     :
                     : "s"(g0), "s"(g1)
                     : "memory");
        asm volatile("s_wait_tensorcnt 0" ::: "memory");
    }
    __syncthreads();                           // data visible to whole block

    int cnt = 0;
#pragma unroll
    for (int j = 0; j < EPT; ++j)              // strided: conflict-free LDS reads
        cnt += (smask[t + j * CT] != 0);
    sc[t] = cnt;
    __syncthreads();
    for (int off = CT / 2; off > 0; off >>= 1) {
        if (t < off) sc[t] += sc[t + off];
        __syncthreads();
    }
    if (t == 0) chunk_cnt[blk] = sc[0];
}

// ---------------------------------------------------------------------------
// K2: serial exclusive scan over chunk counts (2048 entries; trivial work).
// Also emits per-batch valid_num and batch_base. Single-thread = deterministic.
// ---------------------------------------------------------------------------
__global__ void k_scan(const int* __restrict__ chunk_cnt, int* __restrict__ chunk_base,
                       int* __restrict__ valid_num, int* __restrict__ batch_base) {
    if (blockIdx.x != 0 || threadIdx.x != 0) return;
    const int cpb = NCHUNK / B_;               // 64 chunks per batch
    int run = 0;
    for (int b = 0; b < B_; ++b) {
        batch_base[b] = run;
        int vb = 0;
        for (int c = 0; c < cpb; ++c) {
            const int idx = b * cpb + c;
            const int v   = chunk_cnt[idx];
            chunk_base[idx] = run;
            run += v;
            vb  += v;
        }
        valid_num[b] = vb;
    }
}

// ---------------------------------------------------------------------------
// K3: stable compaction. Stage the mask chunk into LDS with async b128 copies
// (each thread stages its own 64B -> only a per-wave s_wait_asynccnt needed
// before the thread reads its own staged region), then block-scan and emit
// flat indices of valid elements in stable row-major order.
// ---------------------------------------------------------------------------
__global__ void k_compact(const int* __restrict__ mask, const int* __restrict__ chunk_base,
                          int* __restrict__ comp_idx) {
    const int blk = blockIdx.x;
    const int t   = threadIdx.x;

    __shared__ int smask[CHUNK];               // 16 KB staged mask chunk
    __shared__ int sscan[CT];

    // Low 32 bits of a flat pointer into LDS aperture == LDS byte address.
    const uint32_t lds_base = (uint32_t)(uintptr_t)(&smask[0]);
    const uint64_t gsrc     = (uint64_t)(uintptr_t)(mask + (size_t)blk * CHUNK) + (uint64_t)t * 64u;
    const uint32_t ldst     = lds_base + (uint32_t)t * 64u;
#pragma unroll
    for (int j = 0; j < 4; ++j)                // 4 x 16B = this thread's 16 ints
        async_copy_b128_to_lds(ldst + (uint32_t)j * 16u, gsrc + (uint64_t)j * 16u);
    wait_asynccnt0();

    const int base_e = t * EPT;
    int cnt = 0;
#pragma unroll
    for (int j = 0; j < EPT; ++j) cnt += (smask[base_e + j] != 0);
    sscan[t] = cnt;
    __syncthreads();

    if (t == 0) {                              // serial exclusive scan, 256 entries
        int run = 0;
        for (int i = 0; i < CT; ++i) { const int c = sscan[i]; sscan[i] = run; run += c; }
    }
    __syncthreads();

    int pos = chunk_base[blk] + sscan[t];
    const int gflat = blk * CHUNK + base_e;
#pragma unroll
    for (int j = 0; j < EPT; ++j) {
        if (smask[base_e + j] != 0) comp_idx[pos++] = gflat + j;
    }
}

// ---------------------------------------------------------------------------
// K4: sampled pairwise loss partials. Each block lies entirely inside one batch.
// ---------------------------------------------------------------------------
__global__ void k_sample(const float* __restrict__ d3, const float* __restrict__ dp,
                         const int* __restrict__ s1r, const int* __restrict__ s2r,
                         const int* __restrict__ comp_idx,
                         const int* __restrict__ valid_num, const int* __restrict__ batch_base,
                         float* __restrict__ psum1, float* __restrict__ psum2,
                         int* __restrict__ pcnt1, int* __restrict__ pcnt2) {
    const int blk = blockIdx.x;
    const int t   = threadIdx.x;
    const int b   = blk >> 3;                  // 1024 samples/block, P=8192 -> 8 blocks/batch
    int vn = valid_num[b];
    if (vn < 1) vn = 1;                        // guard (mask all-zero cannot occur in practice)
    const int bb = batch_base[b];

    float sum1 = 0.f, sum2 = 0.f;
    int   c1 = 0, c2 = 0;
    const int base = blk * (ST * SI);
#pragma unroll
    for (int k = 0; k < SI; ++k) {
        const int gid = base + k * ST + t;
        const int i1  = s1r[gid] % vn + bb;
        const int i2  = s2r[gid] % vn + bb;
        const int f1  = comp_idx[i1];
        const int f2  = comp_idx[i2];
        const float a    = d3[f1];
        const float bv   = d3[f2];
        const float diff = dp[f1] - dp[f2];
        if (a > bv + EPS_) { sum1 += fmaxf(-diff, 0.f); ++c1; }
        if (a < bv - EPS_) { sum2 += fmaxf( diff, 0.f); ++c2; }
    }

    __shared__ float ss1[ST], ss2[ST];
    __shared__ int   sc1[ST], sc2[ST];
    ss1[t] = sum1; ss2[t] = sum2; sc1[t] = c1; sc2[t] = c2;
    __syncthreads();
    for (int off = ST / 2; off > 0; off >>= 1) {   // fixed-order tree: deterministic
        if (t < off) {
            ss1[t] += ss1[t + off]; ss2[t] += ss2[t + off];
            sc1[t] += sc1[t + off]; sc2[t] += sc2[t + off];
        }
        __syncthreads();
    }
    if (t == 0) { psum1[blk] = ss1[0]; psum2[blk] = ss2[0]; pcnt1[blk] = sc1[0]; pcnt2[blk] = sc2[0]; }
}

// ---------------------------------------------------------------------------
// K5: final deterministic combine of SB partials
// ---------------------------------------------------------------------------
__global__ void k_final(const float* __restrict__ psum1, const float* __restrict__ psum2,
                        const int* __restrict__ pcnt1, const int* __restrict__ pcnt2,
                        float* __restrict__ out) {
    const int t = threadIdx.x;
    __shared__ float ss1[SB], ss2[SB];
    __shared__ int   sc1[SB], sc2[SB];
    ss1[t] = psum1[t]; ss2[t] = psum2[t]; sc1[t] = pcnt1[t]; sc2[t] = pcnt2[t];
    __syncthreads();
    for (int off = SB / 2; off > 0; off >>= 1) {
        if (t < off) {
            ss1[t] += ss1[t + off]; ss2[t] += ss2[t + off];
            sc1[t] += sc1[t + off]; sc2[t] += sc2[t + off];
        }
        __syncthreads();
    }
    if (t == 0) {
        const float l1 = ss1[0] / (float)sc1[0];
        const float l2 = ss2[0] / (float)sc2[0];
        out[0] = 0.5f * (l1 + l2);
    }
}

// ---------------------------------------------------------------------------
// Workspace layout (bytes):
//   0        chunk_cnt  int[NCHUNK]   (8 KB)
//   8192     chunk_base int[NCHUNK]   (8 KB)
//   16384    valid_num  int[B_]
//   16512    batch_base int[B_]
//   16640    psum1 float[SB]
//   17664    psum2 float[SB]
//   18688    pcnt1 int[SB]
//   19712    pcnt2 int[SB]
//   32768    comp_idx int[NT_]        (32 MB)
// Total ~33.6 MB.
// ---------------------------------------------------------------------------
extern "C" void kernel_launch(void* const* d_in, const int* in_sizes, int n_in,
                              void* d_out, int out_size, void* d_ws, size_t ws_size,
                              hipStream_t stream) {
    (void)in_sizes; (void)n_in; (void)out_size; (void)ws_size;

    const float* d3   = (const float*)d_in[0];   // depth_3dmm  [B,1,H,W]
    const float* dp   = (const float*)d_in[1];   // depth_pigan [B,1,H,W]
    const int*   mask = (const int*)  d_in[2];   // mask        [B,1,H,W]
    const int*   s1r  = (const int*)  d_in[3];   // sample1_raw [B,P]
    const int*   s2r  = (const int*)  d_in[4];   // sample2_raw [B,P]
    float*       out  = (float*)d_out;

    char* ws = (char*)d_ws;
    int*   chunk_cnt  = (int*)  (ws + 0);
    int*   chunk_base = (int*)  (ws + 8192);
    int*   valid_num  = (int*)  (ws + 16384);
    int*   batch_base = (int*)  (ws + 16512);
    float* psum1      = (float*)(ws + 16640);
    float* psum2      = (float*)(ws + 17664);
    int*   pcnt1      = (int*)  (ws + 18688);
    int*   pcnt2      = (int*)  (ws + 19712);
    int*   comp_idx   = (int*)  (ws + 32768);

    k_count  <<<NCHUNK, CT, 0, stream>>>(mask, chunk_cnt);
    k_scan   <<<1,      32, 0, stream>>>(chunk_cnt, chunk_base, valid_num, batch_base);
    k_compact<<<NCHUNK, CT, 0, stream>>>(mask, chunk_base, comp_idx);
    k_sample <<<SB,     ST, 0, stream>>>(d3, dp, s1r, s2r, comp_idx, valid_num, batch_base,
                                         psum1, psum2, pcnt1, pcnt2);
    k_final  <<<1,      SB, 0, stream>>>(psum1, psum2, pcnt1, pcnt2, out);
}